// CB_RNN_tiedcell_16140487098818
// MI455X (gfx1250) — compile-verified
//
#include <hip/hip_runtime.h>
#include <hip/hip_bf16.h>

typedef __attribute__((ext_vector_type(2))) float v2f;
typedef __attribute__((ext_vector_type(4))) float v4f;
typedef __attribute__((ext_vector_type(8))) float v8f;

#define HDIM  24
#define INDIM 8

__device__ __forceinline__ float softplus_f(float v) {
  return (v > 20.f) ? v : log1pf(__expf(v));
}
__device__ __forceinline__ float sigmoid_f(float v) {
  return __builtin_amdgcn_rcpf(1.f + __expf(-v));
}

// ---------------------------------------------------------------------------
// Stage 1: per-row constants (one tiny block).
// ws layout (floats): [0..191]  P_z (24x8)
//                     [192..215] a_z[h] = (K@r)[h] + b_z[h]
//                     [216..239] a_v[h] = (W@(Ueff*X*r))[h] + b_v[h]
//                     [240..263] v0[h]
// ---------------------------------------------------------------------------
__global__ void cbrnn_precompute(
    const float* __restrict__ W,  const float* __restrict__ Pm,
    const float* __restrict__ bv, const float* __restrict__ bz,
    const float* __restrict__ e,  const float* __restrict__ ep,
    const float* __restrict__ cx, const float* __restrict__ cu,
    const float* __restrict__ cU, const float* __restrict__ v0,
    const float* __restrict__ X0, const float* __restrict__ U0,
    float* __restrict__ ws)
{
  const int h = threadIdx.x;
  if (h >= HDIM) return;
  const float z_min = 0.001f, z_max = 0.1f, delta_t = 1.0f;
  const float spe  = softplus_f(e[0]);
  const float spep = softplus_f(ep[0]);
  float sumK = 0.f, sumV = 0.f;
  for (int j = 0; j < HDIM; ++j) {
    const float rj   = sigmoid_f(v0[j]);
    const float zx   = z_min + (z_max - z_min) * sigmoid_f(cx[j]);
    const float Xj   = zx + (1.f - zx) * X0[j] - delta_t * U0[j] * X0[j] * rj;
    const float zu   = z_min + (z_max - z_min) * sigmoid_f(cu[j]);
    const float Ucap = 0.9f * sigmoid_f(cU[j]);
    float Uj = Ucap * zu + (1.f - zu) * U0[j] + delta_t * Ucap * (1.f - U0[j]) * rj;
    Uj = fminf(fmaxf(Uj, Ucap), 1.f);
    const float Whj = W[h * HDIM + j];
    sumV += Whj * (Uj * Xj * rj);
    sumK += spe * softplus_f(Whj) * rj;
  }
  ws[192 + h] = sumK + bz[h];
  ws[216 + h] = sumV + bv[h];
  ws[240 + h] = v0[h];
  for (int i = 0; i < INDIM; ++i)
    ws[h * INDIM + i] = spep * softplus_f(Pm[h * INDIM + i]);
}

// ---------------------------------------------------------------------------
// Stage 2: batched GEMM (M=24, K=8) + epilogue via V_WMMA_F32_16X16X4_F32.
// One wave per 16-column tile; per tile 8 WMMAs (P and P_z, two M-tiles,
// two K=4 chunks), then sigmoid epilogue, LDS-staged coalesced b128 stores.
// ---------------------------------------------------------------------------
#define WMMA4(A, B, C) \
  __builtin_amdgcn_wmma_f32_16x16x4_f32(false, (A), false, (B), (short)0, (C), false, false)

__global__ __launch_bounds__(256) void cbrnn_main(
    const float* __restrict__ x, const float* __restrict__ Pm,
    const float* __restrict__ ws, float* __restrict__ out,
    int Bsz, int numTiles)
{
  __shared__ __align__(16) float lds[8 * 384];      // 1536 B per wave
  const int lane = threadIdx.x & 31;
  const int wib  = threadIdx.x >> 5;
  const int lh   = lane >> 4;                       // lane half (0/1)
  const int n    = lane & 15;                       // column within tile
  const int m    = n;                               // A row (lanes repeat M)
  const int k0   = 2 * lh;                          // k = k0 + j per A/B layout
  const int wavesTotal = (int)gridDim.x * (int)(blockDim.x >> 5);
  const int waveId     = (int)blockIdx.x * (int)(blockDim.x >> 5) + wib;

  const float* __restrict__ Pz  = ws;
  const float* __restrict__ a_z = ws + 192;
  const float* __restrict__ a_v = ws + 216;
  const float* __restrict__ v0c = ws + 240;

  // ---- A matrices (hoisted, reused across all tiles) ----
  v2f aP0c0, aP0c1, aZ0c0, aZ0c1, aP1c0, aP1c1, aZ1c0, aZ1c1;
  aP0c0[0] = Pm[m*INDIM + k0    ];  aP0c0[1] = Pm[m*INDIM + k0 + 1];
  aP0c1[0] = Pm[m*INDIM + k0 + 4];  aP0c1[1] = Pm[m*INDIM + k0 + 5];
  aZ0c0[0] = Pz[m*INDIM + k0    ];  aZ0c0[1] = Pz[m*INDIM + k0 + 1];
  aZ0c1[0] = Pz[m*INDIM + k0 + 4];  aZ0c1[1] = Pz[m*INDIM + k0 + 5];
  const bool mv = (m < HDIM - 16);                  // rows 16..23 valid
  const int  m1 = 16 + m;
  aP1c0[0] = mv ? Pm[m1*INDIM + k0    ] : 0.f;  aP1c0[1] = mv ? Pm[m1*INDIM + k0 + 1] : 0.f;
  aP1c1[0] = mv ? Pm[m1*INDIM + k0 + 4] : 0.f;  aP1c1[1] = mv ? Pm[m1*INDIM + k0 + 5] : 0.f;
  aZ1c0[0] = mv ? Pz[m1*INDIM + k0    ] : 0.f;  aZ1c0[1] = mv ? Pz[m1*INDIM + k0 + 1] : 0.f;
  aZ1c1[0] = mv ? Pz[m1*INDIM + k0 + 4] : 0.f;  aZ1c1[1] = mv ? Pz[m1*INDIM + k0 + 5] : 0.f;

  // ---- epilogue per-row constants (C/D row m = r + 8*lh; tile1 row 16+r) ----
  float az0[8], av0[8], v00[8], az1[8], av1[8], v01[8];
#pragma unroll
  for (int r = 0; r < 8; ++r) {
    az0[r] = a_z[r + 8*lh]; av0[r] = a_v[r + 8*lh]; v00[r] = v0c[r + 8*lh];
    az1[r] = a_z[16 + r];   av1[r] = a_v[16 + r];   v01[r] = v0c[16 + r];
  }

  float* myLds = lds + wib * 384;

  for (int tile = waveId; tile < numTiles; tile += wavesTotal) {
    const size_t col0 = (size_t)tile * 16;
    const float* xc = x + col0 + n;
    // B matrix 4x16 f32: VGPR j, lane-half lh holds row k = 2*lh + j
    v2f b0, b1;
    b0[0] = xc[(size_t)(k0    ) * Bsz];
    b0[1] = xc[(size_t)(k0 + 1) * Bsz];
    b1[0] = xc[(size_t)(k0 + 4) * Bsz];
    b1[1] = xc[(size_t)(k0 + 5) * Bsz];

    v8f accV0 = {0.f,0.f,0.f,0.f,0.f,0.f,0.f,0.f};
    v8f accV1 = {0.f,0.f,0.f,0.f,0.f,0.f,0.f,0.f};
    v8f accZ0 = {0.f,0.f,0.f,0.f,0.f,0.f,0.f,0.f};
    v8f accZ1 = {0.f,0.f,0.f,0.f,0.f,0.f,0.f,0.f};
    accV0 = WMMA4(aP0c0, b0, accV0);  accV0 = WMMA4(aP0c1, b1, accV0);
    accV1 = WMMA4(aP1c0, b0, accV1);  accV1 = WMMA4(aP1c1, b1, accV1);
    accZ0 = WMMA4(aZ0c0, b0, accZ0);  accZ0 = WMMA4(aZ0c1, b1, accZ0);
    accZ1 = WMMA4(aZ1c0, b0, accZ1);  accZ1 = WMMA4(aZ1c1, b1, accZ1);

    // ---- epilogue: v = (1 - 0.1*sig(z)) * v0 + 0.1*(pv + a_v) ----
    v4f lo, hi;
#pragma unroll
    for (int r = 0; r < 4; ++r) {
      {
        const float zt = 0.1f * sigmoid_f(accZ0[r] + az0[r]);
        lo[r] = (1.f - zt) * v00[r] + 0.1f * (accV0[r] + av0[r]);
      }
      {
        const float zt = 0.1f * sigmoid_f(accZ0[r+4] + az0[r+4]);
        hi[r] = (1.f - zt) * v00[r+4] + 0.1f * (accV0[r+4] + av0[r+4]);
      }
    }
    // LDS tile in (col, h) order -> contiguous (B,H) output region
    *(v4f*)(myLds + n*24 + 8*lh    ) = lo;   // h = 8*lh + 0..3
    *(v4f*)(myLds + n*24 + 8*lh + 4) = hi;   // h = 8*lh + 4..7
    if (lh == 0) {                           // tile1: h = 16..23, rows m<8 only
      v4f lo1, hi1;
#pragma unroll
      for (int r = 0; r < 4; ++r) {
        {
          const float zt = 0.1f * sigmoid_f(accZ1[r] + az1[r]);
          lo1[r] = (1.f - zt) * v01[r] + 0.1f * (accV1[r] + av1[r]);
        }
        {
          const float zt = 0.1f * sigmoid_f(accZ1[r+4] + az1[r+4]);
          hi1[r] = (1.f - zt) * v01[r+4] + 0.1f * (accV1[r+4] + av1[r+4]);
        }
      }
      *(v4f*)(myLds + n*24 + 16) = lo1;
      *(v4f*)(myLds + n*24 + 20) = hi1;
    }
    asm volatile("" ::: "memory");   // keep DS store->load order (HW is in-order per wave)

    // 384 contiguous floats -> 3 coalesced 512B b128 store passes
    float* gdst = out + col0 * 24;
#pragma unroll
    for (int p = 0; p < 3; ++p) {
      const v4f val = *(const v4f*)(myLds + (p*32 + lane) * 4);
      *(v4f*)(gdst + (p*32 + lane) * 4) = val;
    }
    asm volatile("" ::: "memory");   // don't hoist next iter's DS stores above reads
  }
}

extern "C" void kernel_launch(void* const* d_in, const int* in_sizes, int n_in,
                              void* d_out, int out_size, void* d_ws, size_t ws_size,
                              hipStream_t stream) {
  const float* x   = (const float*)d_in[0];
  const float* W   = (const float*)d_in[1];
  const float* P   = (const float*)d_in[2];
  const float* b_v = (const float*)d_in[3];
  const float* b_z = (const float*)d_in[4];
  const float* e   = (const float*)d_in[5];
  const float* e_p = (const float*)d_in[6];
  const float* c_x = (const float*)d_in[7];
  const float* c_u = (const float*)d_in[8];
  const float* c_U = (const float*)d_in[9];
  const float* v0  = (const float*)d_in[10];
  const float* X0  = (const float*)d_in[11];
  const float* U0  = (const float*)d_in[12];
  float* out = (float*)d_out;
  float* ws  = (float*)d_ws;

  const int Bsz   = in_sizes[0] / INDIM;   // x is (IN, B)
  const int tiles = Bsz / 16;              // B is a multiple of 16

  cbrnn_precompute<<<1, 32, 0, stream>>>(W, P, b_v, b_z, e, e_p,
                                         c_x, c_u, c_U, v0, X0, U0, ws);

  int blocks = (tiles + 7) / 8;            // 8 waves per 256-thread block
  if (blocks > 2048) blocks = 2048;        // grid-stride beyond this
  if (blocks < 1) blocks = 1;
  cbrnn_main<<<blocks, 256, 0, stream>>>(x, P, ws, out, Bsz, tiles);
}